// DGCNN_9268539424869
// MI455X (gfx1250) — compile-verified
//
#include <hip/hip_runtime.h>
#include <hip/hip_bf16.h>

// DGCNN forward, restructured:
//   edge conv  h[c,n,k] = Y1[c, idx[n,k]] + Z[c,n],  Y1 = W1*X, Z = (W2-W1)*X
// All GEMMs run on V_WMMA_F32_16X16X4_F32 (exact f32 math -> identical kNN
// ordering to the f32 reference). K dims padded to multiples of 8; the WMMA
// inner loops are guard-free and software-pipelined (ping-pong fragment
// buffers) so next-step loads overlap the current WMMA pack.

#define NPTS   2048
#define BATCH  4
#define K_NBR  20
#define EPSV   1e-5f

typedef float v2f __attribute__((ext_vector_type(2)));
typedef float v8f __attribute__((ext_vector_type(8)));

struct Frag { v2f a; v2f b[4]; };

__device__ __forceinline__ void do_mma(const Frag& f, v8f acc[4]) {
#pragma unroll
    for (int s = 0; s < 4; ++s)
        acc[s] = __builtin_amdgcn_wmma_f32_16x16x4_f32(false, f.a, false, f.b[s],
                                                       (short)0, acc[s], false, false);
}

// ---------------------------------------------------------------- row norms
__global__ __launch_bounds__(256)
void rownorm_kernel(const float* __restrict__ X, long bstride, int D, int N,
                    float* __restrict__ xx) {
    int t = blockIdx.x * blockDim.x + threadIdx.x;
    if (t >= BATCH * N) return;
    int b = t / N, n = t % N;
    const float* Xb = X + (size_t)b * bstride;
    float s = 0.f;
    for (int d = 0; d < D; ++d) { float v = Xb[(size_t)d * N + n]; s += v * v; }
    xx[t] = s;
}

// ------------------------------------------------- pad block-1 input 6 -> 8
__global__ __launch_bounds__(256)
void pad_kernel(const float* __restrict__ X, float* __restrict__ Xp, int N) {
    int t = blockIdx.x * blockDim.x + threadIdx.x;
    if (t >= BATCH * 8 * N) return;
    int b = t / (8 * N);
    int r = t % (8 * N);
    int d = r / N, n = r % N;
    Xp[t] = (d < 6) ? X[((size_t)b * 6 + d) * N + n] : 0.f;
}

// --------------------------------------------------- pairwise distance (WMMA)
// dist[b][n][m] = xx[n] + xx[m] - 2 * sum_d X[d][n] * X[d][m]
// One wave per 16(n) x 64(m) strip; D multiple of 8 (padded).
__device__ __forceinline__ Frag gram_load(const float* __restrict__ Xb, int N,
                                          int ka, int na, int mcol) {
    Frag f;
    f.a.x = Xb[(size_t)ka * N + na];
    f.a.y = Xb[(size_t)(ka + 1) * N + na];
#pragma unroll
    for (int s = 0; s < 4; ++s) {
        f.b[s].x = Xb[(size_t)ka * N + mcol + 16 * s];
        f.b[s].y = Xb[(size_t)(ka + 1) * N + mcol + 16 * s];
    }
    return f;
}

__global__ __launch_bounds__(32)
void gram_kernel(const float* __restrict__ X, long bstride,
                 const float* __restrict__ xx, float* __restrict__ dist,
                 int D, int N) {
    int b     = blockIdx.z;
    int n0    = blockIdx.y * 16;
    int mbase = blockIdx.x * 64;
    int lane  = threadIdx.x;
    int half  = lane >> 4;
    int lr    = lane & 15;
    const float* Xb = X + (size_t)b * bstride;
    int na   = n0 + lr;
    int mcol = mbase + lr;
    int koff = 2 * half;

    v8f acc[4] = {{}, {}, {}, {}};
    int S = D / 4;                 // even (D % 8 == 0)
    Frag f0 = gram_load(Xb, N, koff, na, mcol);
    Frag f1;
    int st = 0;
    for (; st + 2 < S; st += 2) {
        f1 = gram_load(Xb, N, (st + 1) * 4 + koff, na, mcol);
        do_mma(f0, acc);
        f0 = gram_load(Xb, N, (st + 2) * 4 + koff, na, mcol);
        do_mma(f1, acc);
    }
    f1 = gram_load(Xb, N, (st + 1) * 4 + koff, na, mcol);
    do_mma(f0, acc);
    do_mma(f1, acc);

    const float* xxb  = xx + (size_t)b * N;
    float*       base = dist + ((size_t)b * N + n0) * N;
    float xm0 = xxb[mcol];
    float xm1 = xxb[mcol + 16];
    float xm2 = xxb[mcol + 32];
    float xm3 = xxb[mcol + 48];
#pragma unroll
    for (int j = 0; j < 8; ++j) {
        int   row = j + 8 * half;
        float xn  = xxb[n0 + row];
        float* dr = base + (size_t)row * N + mbase + lr;
        dr[0]  = xn + xm0 - 2.0f * acc[0][j];
        dr[16] = xn + xm1 - 2.0f * acc[1][j];
        dr[32] = xn + xm2 - 2.0f * acc[2][j];
        dr[48] = xn + xm3 - 2.0f * acc[3][j];
    }
}

// ---------------------------------------------------------------- top-k(20)
__global__ __launch_bounds__(128)
void topk_kernel(const float* __restrict__ dist, int* __restrict__ idx, int N) {
    int t = blockIdx.x * blockDim.x + threadIdx.x;
    if (t >= BATCH * N) return;
    int b = t / N, n = t % N;
    const float* row = dist + ((size_t)b * N + n) * N;
    float best[K_NBR]; int bi[K_NBR];
#pragma unroll
    for (int k = 0; k < K_NBR; ++k) { best[k] = 3.4e38f; bi[k] = 0; }
    float worst = 3.4e38f; int wpos = 0;
    for (int m = 0; m < N; ++m) {
        float d = row[m];
        if (d < worst) {
            best[wpos] = d; bi[wpos] = m;
            worst = -3.4e38f;
#pragma unroll
            for (int k = 0; k < K_NBR; ++k)
                if (best[k] > worst) { worst = best[k]; wpos = k; }
        }
    }
    int* out = idx + ((size_t)b * N + n) * K_NBR;
#pragma unroll
    for (int k = 0; k < K_NBR; ++k) out[k] = bi[k];
}

// --------------------------------------------------------- weight preprocess
// W is (Cout, 2*Cin). Wp is (2*Cout, Kpad): rows [0,Cout)=W1, rows
// [Cout,2Cout)=W2-W1, columns >= Cin zero-filled.
__global__ __launch_bounds__(256)
void wprep_kernel(const float* __restrict__ W, float* __restrict__ Wp,
                  int Cout, int Cin, int Kpad) {
    int t = blockIdx.x * blockDim.x + threadIdx.x;
    if (t >= Cout * Kpad) return;
    int c = t / Kpad, d = t % Kpad;
    float w1 = 0.f, w2 = 0.f;
    if (d < Cin) {
        w1 = W[(size_t)c * (2 * Cin) + d];
        w2 = W[(size_t)c * (2 * Cin) + Cin + d];
    }
    Wp[(size_t)c * Kpad + d]          = w1;
    Wp[(size_t)(Cout + c) * Kpad + d] = w2 - w1;
}

// --------------------------------------------------- generic batched GEMM
// Out[b][m][n] = sum_k A[m][k] * Bmat[b][k][n]; Kd multiple of 8, no guards.
// A-fragment = one aligned float2 load (adjacent K in row-major A).
__device__ __forceinline__ Frag gemm_load(const float* __restrict__ A,
                                          const float* __restrict__ Bb,
                                          int Kd, int N, int ka, int ma, int ncol) {
    Frag f;
    f.a = *(const v2f*)(A + (size_t)ma * Kd + ka);
#pragma unroll
    for (int s = 0; s < 4; ++s) {
        f.b[s].x = Bb[(size_t)ka * N + ncol + 16 * s];
        f.b[s].y = Bb[(size_t)(ka + 1) * N + ncol + 16 * s];
    }
    return f;
}

__global__ __launch_bounds__(32)
void gemm_kernel(const float* __restrict__ A, const float* __restrict__ Bmat,
                 float* __restrict__ Out, int M, int Kd, int N,
                 long bStrideB, long bStrideO) {
    int b     = blockIdx.z;
    int m0    = blockIdx.y * 16;
    int nbase = blockIdx.x * 64;
    int lane  = threadIdx.x;
    int half  = lane >> 4;
    int lr    = lane & 15;
    const float* Bb = Bmat + (size_t)b * bStrideB;
    int ma   = m0 + lr;
    int ncol = nbase + lr;
    int koff = 2 * half;

    v8f acc[4] = {{}, {}, {}, {}};
    int S = Kd / 4;                // even (Kd % 8 == 0)
    Frag f0 = gemm_load(A, Bb, Kd, N, koff, ma, ncol);
    Frag f1;
    int st = 0;
    for (; st + 2 < S; st += 2) {
        f1 = gemm_load(A, Bb, Kd, N, (st + 1) * 4 + koff, ma, ncol);
        do_mma(f0, acc);
        f0 = gemm_load(A, Bb, Kd, N, (st + 2) * 4 + koff, ma, ncol);
        do_mma(f1, acc);
    }
    f1 = gemm_load(A, Bb, Kd, N, (st + 1) * 4 + koff, ma, ncol);
    do_mma(f0, acc);
    do_mma(f1, acc);

    float* Ob = Out + (size_t)b * bStrideO;
#pragma unroll
    for (int j = 0; j < 8; ++j) {
        int    row  = j + 8 * half;
        float* orow = Ob + (size_t)(m0 + row) * N + nbase + lr;
        orow[0]  = acc[0][j];
        orow[16] = acc[1][j];
        orow[32] = acc[2][j];
        orow[48] = acc[3][j];
    }
}

// ------------------------------------------------------------------- zeroing
__global__ void zero_kernel(float* __restrict__ p, int n) {
    int t = blockIdx.x * blockDim.x + threadIdx.x;
    if (t < n) p[t] = 0.f;
}

// --------------------------------------- edge gather + stats + k-extremes
__global__ __launch_bounds__(256)
void edge_stats_kernel(const float* __restrict__ Y, const int* __restrict__ idx,
                       float* __restrict__ vmax, float* __restrict__ vmin,
                       float* __restrict__ gacc, int Cout, int N, int groups) {
    int bc = blockIdx.x;
    int b = bc / Cout, c = bc % Cout;
    const float* Y1   = Y + ((size_t)b * 2 * Cout + c) * N;
    const float* Zr   = Y + ((size_t)b * 2 * Cout + Cout + c) * N;
    const int*   idxb = idx + (size_t)b * N * K_NBR;
    float s1 = 0.f, s2 = 0.f;
    for (int n = threadIdx.x; n < N; n += blockDim.x) {
        float z = Zr[n];
        const int* in = idxb + (size_t)n * K_NBR;
        float mx = -3.4e38f, mn = 3.4e38f;
#pragma unroll
        for (int k = 0; k < K_NBR; ++k) {
            float v = Y1[in[k]] + z;
            s1 += v; s2 += v * v;
            mx = fmaxf(mx, v); mn = fminf(mn, v);
        }
        vmax[((size_t)b * Cout + c) * N + n] = mx;
        vmin[((size_t)b * Cout + c) * N + n] = mn;
    }
    __shared__ float ls1[256], ls2[256];
    ls1[threadIdx.x] = s1; ls2[threadIdx.x] = s2;
    __syncthreads();
    for (int off = 128; off > 0; off >>= 1) {
        if (threadIdx.x < off) {
            ls1[threadIdx.x] += ls1[threadIdx.x + off];
            ls2[threadIdx.x] += ls2[threadIdx.x + off];
        }
        __syncthreads();
    }
    if (threadIdx.x == 0) {
        int g = c / (Cout / groups);
        atomicAdd(&gacc[((size_t)b * groups + g) * 2 + 0], ls1[0]);
        atomicAdd(&gacc[((size_t)b * groups + g) * 2 + 1], ls2[0]);
    }
}

// ------------------------------------ GroupNorm+leaky+max_k -> hcat channel
__global__ __launch_bounds__(256)
void edge_final_kernel(const float* __restrict__ vmax, const float* __restrict__ vmin,
                       const float* __restrict__ gacc,
                       const float* __restrict__ gamma, const float* __restrict__ beta,
                       float* __restrict__ outX, int Cout, int N, int groups,
                       int chanStride, int chanOff) {
    int t = blockIdx.x * blockDim.x + threadIdx.x;
    int total = BATCH * Cout * N;
    if (t >= total) return;
    int b = t / (Cout * N);
    int r = t % (Cout * N);
    int c = r / N, n = r % N;
    int g = c / (Cout / groups);
    float cnt = (float)(Cout / groups) * (float)N * (float)K_NBR;
    float S1 = gacc[((size_t)b * groups + g) * 2 + 0];
    float S2 = gacc[((size_t)b * groups + g) * 2 + 1];
    float mu  = S1 / cnt;
    float var = S2 / cnt - mu * mu;
    float inv = rsqrtf(var + EPSV);
    float s  = gamma[c] * inv;
    float tt = beta[c] - mu * s;
    size_t off = ((size_t)b * Cout + c) * N + n;
    float ex = (s >= 0.f) ? vmax[off] : vmin[off];
    float o = s * ex + tt;
    o = (o >= 0.f) ? o : 0.2f * o;
    outX[((size_t)b * chanStride + chanOff + c) * N + n] = o;
}

// --------------------------------------------- final GroupNorm stats (no k)
__global__ __launch_bounds__(256)
void plain_stats_kernel(const float* __restrict__ H, float* __restrict__ gacc,
                        int Cout, int N, int groups) {
    int bc = blockIdx.x;
    int b = bc / Cout, c = bc % Cout;
    const float* Hr = H + ((size_t)b * Cout + c) * N;
    float s1 = 0.f, s2 = 0.f;
    for (int n = threadIdx.x; n < N; n += blockDim.x) {
        float v = Hr[n];
        s1 += v; s2 += v * v;
    }
    __shared__ float ls1[256], ls2[256];
    ls1[threadIdx.x] = s1; ls2[threadIdx.x] = s2;
    __syncthreads();
    for (int off = 128; off > 0; off >>= 1) {
        if (threadIdx.x < off) {
            ls1[threadIdx.x] += ls1[threadIdx.x + off];
            ls2[threadIdx.x] += ls2[threadIdx.x + off];
        }
        __syncthreads();
    }
    if (threadIdx.x == 0) {
        int g = c / (Cout / groups);
        atomicAdd(&gacc[((size_t)b * groups + g) * 2 + 0], ls1[0]);
        atomicAdd(&gacc[((size_t)b * groups + g) * 2 + 1], ls2[0]);
    }
}

// ---------------------- final affine+leaky then global max & mean over n
__global__ __launch_bounds__(256)
void final_out_kernel(const float* __restrict__ H, const float* __restrict__ gacc,
                      const float* __restrict__ gamma, const float* __restrict__ beta,
                      float* __restrict__ out, int Cout, int N, int groups) {
    int bc = blockIdx.x;
    int b = bc / Cout, c = bc % Cout;
    int g = c / (Cout / groups);
    float cnt = (float)(Cout / groups) * (float)N;
    float S1 = gacc[((size_t)b * groups + g) * 2 + 0];
    float S2 = gacc[((size_t)b * groups + g) * 2 + 1];
    float mu  = S1 / cnt;
    float var = S2 / cnt - mu * mu;
    float inv = rsqrtf(var + EPSV);
    float s  = gamma[c] * inv;
    float tt = beta[c] - mu * s;
    const float* Hr = H + ((size_t)b * Cout + c) * N;
    float mx = -3.4e38f, sm = 0.f;
    for (int n = threadIdx.x; n < N; n += blockDim.x) {
        float o = s * Hr[n] + tt;
        o = (o >= 0.f) ? o : 0.2f * o;
        mx = fmaxf(mx, o); sm += o;
    }
    __shared__ float lmx[256], lsm[256];
    lmx[threadIdx.x] = mx; lsm[threadIdx.x] = sm;
    __syncthreads();
    for (int off = 128; off > 0; off >>= 1) {
        if (threadIdx.x < off) {
            lmx[threadIdx.x] = fmaxf(lmx[threadIdx.x], lmx[threadIdx.x + off]);
            lsm[threadIdx.x] += lsm[threadIdx.x + off];
        }
        __syncthreads();
    }
    if (threadIdx.x == 0) {
        out[(size_t)b * (2 * Cout) + c]        = lmx[0];
        out[(size_t)b * (2 * Cout) + Cout + c] = lsm[0] / (float)N;
    }
}

// =========================================================================
// Kpad = padded K (multiple of 8); Cin = logical input channels.
static void run_edge_block(const float* Xin, long xBatchStride, int Cin, int Kpad,
                           const float* W, const float* gamma, const float* beta,
                           int Cout, int groups, int chanOff,
                           float* xx, float* dist, int* idx, float* Y, float* Wp,
                           float* hcat, float* vmx, float* vmn, float* gacc,
                           hipStream_t stream) {
    const int N = NPTS, B = BATCH;
    rownorm_kernel<<<(B * N + 255) / 256, 256, 0, stream>>>(Xin, xBatchStride, Kpad, N, xx);
    gram_kernel<<<dim3(N / 64, N / 16, B), 32, 0, stream>>>(Xin, xBatchStride, xx, dist, Kpad, N);
    topk_kernel<<<(B * N + 127) / 128, 128, 0, stream>>>(dist, idx, N);
    wprep_kernel<<<(Cout * Kpad + 255) / 256, 256, 0, stream>>>(W, Wp, Cout, Cin, Kpad);
    gemm_kernel<<<dim3(N / 64, (2 * Cout) / 16, B), 32, 0, stream>>>(
        Wp, Xin, Y, 2 * Cout, Kpad, N, xBatchStride, (long)2 * Cout * N);
    zero_kernel<<<1, 256, 0, stream>>>(gacc, B * groups * 2);
    edge_stats_kernel<<<B * Cout, 256, 0, stream>>>(Y, idx, vmx, vmn, gacc, Cout, N, groups);
    edge_final_kernel<<<(B * Cout * N + 255) / 256, 256, 0, stream>>>(
        vmx, vmn, gacc, gamma, beta, hcat, Cout, N, groups, 960, chanOff);
}

extern "C" void kernel_launch(void* const* d_in, const int* in_sizes, int n_in,
                              void* d_out, int out_size, void* d_ws, size_t ws_size,
                              hipStream_t stream) {
    const int N = NPTS, B = BATCH;
    const float* x  = (const float*)d_in[0];
    const float* w1 = (const float*)d_in[1];
    const float* g1 = (const float*)d_in[2];
    const float* b1 = (const float*)d_in[3];
    const float* w2 = (const float*)d_in[4];
    const float* g2 = (const float*)d_in[5];
    const float* b2 = (const float*)d_in[6];
    const float* w3 = (const float*)d_in[7];
    const float* g3 = (const float*)d_in[8];
    const float* b3 = (const float*)d_in[9];
    const float* w4 = (const float*)d_in[10];
    const float* g4 = (const float*)d_in[11];
    const float* b4 = (const float*)d_in[12];
    const float* wa = (const float*)d_in[13];
    const float* ga = (const float*)d_in[14];
    const float* ba = (const float*)d_in[15];

    // workspace carve-up (floats)
    float* ws = (float*)d_ws;
    size_t off = 0;
    float* xx   = ws + off; off += (size_t)B * N;
    float* dist = ws + off; off += (size_t)B * N * N;        // 64 MB (L2-resident)
    int*   idx  = (int*)(ws + off); off += (size_t)B * N * K_NBR;
    float* Y    = ws + off; off += (size_t)B * 1024 * N;     // [Y1;Z]
    float* Wp   = ws + off; off += (size_t)2 * 512 * 512;
    float* hcat = ws + off; off += (size_t)B * 960 * N;      // concat(x1..x4)
    float* vmx  = ws + off; off += (size_t)B * 512 * N;
    float* vmn  = ws + off; off += (size_t)B * 512 * N;
    float* gacc = ws + off; off += 256;
    float* h2   = ws + off; off += (size_t)B * 512 * N;
    float* xpad = ws + off; off += (size_t)B * 8 * N;        // block-1 padded input

    // block 1 input: zero-pad 6 -> 8 channels so WMMA loops are guard-free
    pad_kernel<<<(B * 8 * N + 255) / 256, 256, 0, stream>>>(x, xpad, N);

    run_edge_block(xpad,                   (long)8 * N,   6,   8,   w1, g1, b1,  64,  8,   0, xx, dist, idx, Y, Wp, hcat, vmx, vmn, gacc, stream);
    run_edge_block(hcat + (size_t)0 * N,   (long)960 * N, 64,  64,  w2, g2, b2, 128,  8,  64, xx, dist, idx, Y, Wp, hcat, vmx, vmn, gacc, stream);
    run_edge_block(hcat + (size_t)64 * N,  (long)960 * N, 128, 128, w3, g3, b3, 256, 16, 192, xx, dist, idx, Y, Wp, hcat, vmx, vmn, gacc, stream);
    run_edge_block(hcat + (size_t)192 * N, (long)960 * N, 256, 256, w4, g4, b4, 512, 32, 448, xx, dist, idx, Y, Wp, hcat, vmx, vmn, gacc, stream);

    // final 960 -> 512 pointwise conv (WMMA GEMM), GN stats, pooled output
    gemm_kernel<<<dim3(N / 64, 512 / 16, B), 32, 0, stream>>>(
        wa, hcat, h2, 512, 960, N, (long)960 * N, (long)512 * N);
    zero_kernel<<<1, 256, 0, stream>>>(gacc, B * 32 * 2);
    plain_stats_kernel<<<B * 512, 256, 0, stream>>>(h2, gacc, 512, N, 32);
    final_out_kernel<<<B * 512, 256, 0, stream>>>(h2, gacc, ga, ba, (float*)d_out, 512, N, 32);
}